// TripletLoss_3410204033331
// MI455X (gfx1250) — compile-verified
//
#include <hip/hip_runtime.h>
#include <math.h>

// Triplet loss (batch-hard) fused kernel for MI455X (gfx1250, wave32).
// Gram via chained V_WMMA_F32_16X16X4_F32; B tiles double-buffered in LDS
// (GLOBAL_LOAD_ASYNC_TO_LDS_B128 when available); fused masked row max/min.
//
// WMMA f32 16x16x4 layout (CDNA5 ISA 7.12.2):
//   A (16x4):  lane = m + 16*kh holds A[m][2*kh + {0,1}]
//   B (4x16):  lane = n + 16*kh holds B[2*kh + {0,1}][n]
//   C/D:       VGPR v, lane l holds C[v + 8*(l>>4)][l & 15]

typedef __attribute__((ext_vector_type(2))) float v2f;
typedef __attribute__((ext_vector_type(8))) float v8f;
typedef __attribute__((__vector_size__(4 * sizeof(int)))) int v4i;

#define D 128
#define LDS_STRIDE 132   // 128 + 4 pad: ds_load_b64 fragment reads hit all 64 banks once
#define MARGIN 0.3f
#define NEG_FILL 1000000.0f

#if defined(__AMDGCN__) && __has_builtin(__builtin_amdgcn_global_load_async_to_lds_b128)
#define TL_ASYNC 1
#else
#define TL_ASYNC 0
#endif

__device__ __forceinline__ void tl_wait_async() {
#if TL_ASYNC
#if __has_builtin(__builtin_amdgcn_s_wait_asynccnt)
  __builtin_amdgcn_s_wait_asynccnt(0);
#else
  asm volatile("s_wait_asynccnt 0" ::: "memory");
#endif
#endif
}

#if TL_ASYNC
// Copy one 16x128 fp32 tile (rows j0..j0+15) into LDS buf, 16B per lane per op.
__device__ __forceinline__ void tl_stage_async(const float* __restrict__ F,
                                               int j0, float* buf, int tid) {
#pragma unroll
  for (int it = 0; it < 2; ++it) {
    int q   = tid + it * 256;   // float4 index 0..511 (16 rows x 32 float4)
    int row = q >> 5;
    int c4  = q & 31;
    const float* g = F + (size_t)(j0 + row) * D + c4 * 4;
    float* l = buf + row * LDS_STRIDE + c4 * 4;
    __builtin_amdgcn_global_load_async_to_lds_b128(
        (__attribute__((address_space(1))) v4i*)g,
        (__attribute__((address_space(3))) v4i*)l, 0, 0);
  }
}
#endif

// ---------------------------------------------------------------- row norms
__global__ void tl_rownorm(const float* __restrict__ F, float* __restrict__ sq,
                           int N) {
  int r = blockIdx.x * blockDim.x + threadIdx.x;
  if (r >= N) return;
  const float4* p = (const float4*)(F + (size_t)r * D);
  float acc = 0.f;
#pragma unroll
  for (int kk = 0; kk < D / 4; ++kk) {
    float4 x = p[kk];
    acc += x.x * x.x + x.y * x.y + x.z * x.z + x.w * x.w;
  }
  sq[r] = acc;
}

// ------------------------------------------------------------- main fused
// blockDim = 256 (8 waves); each wave owns a 16-row strip; the block shares
// LDS-staged B tiles across its waves.
__global__ void __launch_bounds__(256)
tl_main(const float* __restrict__ F, const long long* __restrict__ labels,
        const float* __restrict__ sq, float* __restrict__ loss, int N) {
  __shared__ float smem[2][16 * LDS_STRIDE];
  const int tid  = threadIdx.x;
  const int wave = tid >> 5;
  const int lane = tid & 31;
  const int i0 = (blockIdx.x * 8 + wave) * 16;  // this wave's 16-row strip
  const int n  = lane & 15;
  const int kh = lane >> 4;

  // Preload A fragments (16x128 strip -> 64 VGPRs).
  const float* Arow = F + (size_t)(i0 + n) * D + 2 * kh;
  v2f a[32];
#pragma unroll
  for (int kk = 0; kk < 32; ++kk) a[kk] = *(const v2f*)(Arow + 4 * kk);

  float sqr[8];
  long long lr[8];
#pragma unroll
  for (int v = 0; v < 8; ++v) {
    int r = i0 + v + 8 * kh;
    sqr[v] = sq[r];
    lr[v]  = labels[r];
  }

  float maxpos[8], minneg[8];
#pragma unroll
  for (int v = 0; v < 8; ++v) {
    maxpos[v] = -__builtin_inff();
    minneg[v] =  __builtin_inff();
  }

  const int T = N / 16;

  // Prologue: stage tile 0 into smem[0].
#if TL_ASYNC
  tl_stage_async(F, 0, smem[0], tid);
  tl_wait_async();
#else
#pragma unroll
  for (int it = 0; it < 2; ++it) {
    int q = tid + it * 256, row = q >> 5, c4 = q & 31;
    float4 x = *(const float4*)(F + (size_t)row * D + c4 * 4);
    *(float4*)(&smem[0][row * LDS_STRIDE + c4 * 4]) = x;
  }
#endif
  __syncthreads();

  for (int t = 0; t < T; ++t) {
    const int j0 = t * 16;
    const float* bufc = smem[t & 1];
    const bool have_next = (t + 1 < T);

#if TL_ASYNC
    if (have_next) tl_stage_async(F, j0 + 16, smem[(t + 1) & 1], tid);
#else
    float4 p0 = {}, p1 = {};
    if (have_next) {
      int q0 = tid,        r0 = q0 >> 5, c0 = q0 & 31;
      int q1 = tid + 256,  r1 = q1 >> 5, c1 = q1 & 31;
      p0 = *(const float4*)(F + (size_t)(j0 + 16 + r0) * D + c0 * 4);
      p1 = *(const float4*)(F + (size_t)(j0 + 16 + r1) * D + c1 * 4);
    }
#endif

    // Batched fragment fetch from LDS (conflict-free), then the WMMA chain.
    const float* Bl = bufc + n * LDS_STRIDE + 2 * kh;
    v2f b[32];
#pragma unroll
    for (int kk = 0; kk < 32; ++kk) b[kk] = *(const v2f*)(Bl + 4 * kk);

    v8f c = {0.f, 0.f, 0.f, 0.f, 0.f, 0.f, 0.f, 0.f};
#pragma unroll
    for (int kk = 0; kk < 32; ++kk)
      c = __builtin_amdgcn_wmma_f32_16x16x4_f32(
          false, a[kk], false, b[kk], (short)0, c, false, false);

    const int cidx = j0 + n;
    const float sqc = sq[cidx];
    const long long lc = labels[cidx];
#pragma unroll
    for (int v = 0; v < 8; ++v) {
      int r = i0 + v + 8 * kh;
      float d2 = sqr[v] + sqc - 2.0f * c[v];
      float dist = d2 > 0.f ? __builtin_sqrtf(d2) : 0.f;
      if (lr[v] == lc) {
        if (r != cidx) maxpos[v] = fmaxf(maxpos[v], dist);
      } else {
        minneg[v] = fminf(minneg[v], dist);
      }
    }

#if !TL_ASYNC
    if (have_next) {
      float* nb = (float*)smem[(t + 1) & 1];
      int q0 = tid,       r0 = q0 >> 5, c0 = q0 & 31;
      int q1 = tid + 256, r1 = q1 >> 5, c1 = q1 & 31;
      *(float4*)(&nb[r0 * LDS_STRIDE + c0 * 4]) = p0;
      *(float4*)(&nb[r1 * LDS_STRIDE + c1 * 4]) = p1;
    }
#endif
    tl_wait_async();
    __syncthreads();
  }

  // Cross-lane reduction within each 16-lane half (xor masks < 16 never mix
  // halves, matching the C/D row split).
#pragma unroll
  for (int m = 8; m >= 1; m >>= 1) {
#pragma unroll
    for (int v = 0; v < 8; ++v) {
      maxpos[v] = fmaxf(maxpos[v], __shfl_xor(maxpos[v], m, 32));
      minneg[v] = fminf(minneg[v], __shfl_xor(minneg[v], m, 32));
    }
  }

  if (n == 0) {
#pragma unroll
    for (int v = 0; v < 8; ++v) {
      int r = i0 + v + 8 * kh;
      float ap = (maxpos[v] == -__builtin_inff()) ? 0.f : maxpos[v];
      float an = (minneg[v] ==  __builtin_inff()) ? NEG_FILL : minneg[v];
      loss[r] = fmaxf(0.f, MARGIN - (an - ap));
    }
  }
}

// ------------------------------------------------------- deterministic sum
__global__ void tl_reduce(const float* __restrict__ loss, float* __restrict__ out,
                          int N) {
  __shared__ float s[256];
  float acc = 0.f;
  for (int i = threadIdx.x; i < N; i += 256) acc += loss[i];
  s[threadIdx.x] = acc;
  __syncthreads();
#pragma unroll
  for (int stride = 128; stride >= 1; stride >>= 1) {
    if (threadIdx.x < stride) s[threadIdx.x] += s[threadIdx.x + stride];
    __syncthreads();
  }
  if (threadIdx.x == 0) out[0] = s[0] / (float)N;
}

extern "C" void kernel_launch(void* const* d_in, const int* in_sizes, int n_in,
                              void* d_out, int out_size, void* d_ws, size_t ws_size,
                              hipStream_t stream) {
  const float* F = (const float*)d_in[0];
  const long long* labels = (const long long*)d_in[1];  // reference dtype int64
  float* out = (float*)d_out;

  const int N = in_sizes[1];  // 8192
  float* sq   = (float*)d_ws;
  float* loss = sq + N;

  tl_rownorm<<<(N + 255) / 256, 256, 0, stream>>>(F, sq, N);
  tl_main<<<N / 128, 256, 0, stream>>>(F, labels, sq, loss, N);
  tl_reduce<<<1, 256, 0, stream>>>(loss, out, N);
}